// GCNNet_50869592653866
// MI455X (gfx1250) — compile-verified
//
#include <hip/hip_runtime.h>
#include <hip/hip_bf16.h>

typedef __attribute__((ext_vector_type(2))) float v2f;
typedef __attribute__((ext_vector_type(8))) float v8f;

#define NN   100000   // nodes
#define KD   512      // in_dim
#define HD   16       // hidden
#define OD   7        // out
#define NTILE 6250    // NN/16

// ---------------------------------------------------------------------------
// zero h1 [NN*HD] and out [NN*OD]
// ---------------------------------------------------------------------------
__global__ __launch_bounds__(256) void zero_kernel(float* __restrict__ h1,
                                                   float* __restrict__ out,
                                                   int n1, int n2) {
  int i = blockIdx.x * 256 + threadIdx.x;
  if (i < n1) h1[i] = 0.0f;
  if (i < n2) out[i] = 0.0f;
}

// ---------------------------------------------------------------------------
// GEMM1: XW[n][16] = (X[n][:] * M1[n][:]) @ W1[512][16] + b1
// 8 waves/block, one 16-row tile per wave, K stepped 32 at a time via LDS,
// 8x v_wmma_f32_16x16x4_f32 per K-chunk.
// Staging LDS is wave-private: DS ops are in-order per wave (CDNA5 ISA), so
// only a compiler scheduling fence (wave_barrier) is needed, no s_barrier.
// ---------------------------------------------------------------------------
__global__ __launch_bounds__(256) void gemm1_kernel(const float* __restrict__ X,
                                                    const float* __restrict__ M1,
                                                    const float* __restrict__ W1,
                                                    const float* __restrict__ b1,
                                                    float* __restrict__ XW) {
  __shared__ float sWt[HD * 516];       // W1 transposed [n][k], pad 516 -> conflict-free
  __shared__ float sA[8][16 * 36];      // per-wave 16x32 chunk, row stride 36 (16B-aligned)

  const int t    = threadIdx.x;
  const int wave = t >> 5;
  const int lane = t & 31;

  // load + transpose W1 into LDS (one time, block-shared -> real barrier)
  for (int i = t; i < KD * HD; i += 256) {
    int k = i >> 4, n = i & 15;
    sWt[n * 516 + k] = W1[i];
  }
  __syncthreads();

  int tile = blockIdx.x * 8 + wave;
  if (tile >= NTILE) tile = NTILE - 1;           // duplicate work, identical stores

  const int g  = lane >> 4;        // K-pair group (0: K{0,1}, 1: K{2,3})
  const int m  = lane & 15;        // matrix row (A) / col (B)
  const int r  = lane >> 1;        // staging row
  const int ch = (lane & 1) * 16;  // staging col half
  const size_t rowbase = (size_t)tile * 16;

  float* sAw = &sA[wave][0];
  const float* wrow = &sWt[m * 516];

  v8f acc = {};

  for (int k0 = 0; k0 < KD; k0 += 32) {
    // stage 16x32 chunk of X*mask (coalesced float4)
    const float* xp = X  + (rowbase + r) * KD + k0 + ch;
    const float* mp = M1 + (rowbase + r) * KD + k0 + ch;
#pragma unroll
    for (int u = 0; u < 16; u += 4) {
      float4 xv = *(const float4*)(xp + u);
      float4 mv = *(const float4*)(mp + u);
      float4 pv = make_float4(xv.x * mv.x, xv.y * mv.y, xv.z * mv.z, xv.w * mv.w);
      *(float4*)&sAw[r * 36 + ch + u] = pv;
    }
    __builtin_amdgcn_wave_barrier();   // order stores vs. loads (HW DS is in-order per wave)
#pragma unroll
    for (int kk = 0; kk < 32; kk += 4) {
      // A frag: A[m][kk + 2g + {0,1}]   (32-bit A 16x4 layout)
      v2f a = *(const v2f*)&sAw[m * 36 + kk + 2 * g];
      // B frag: W1[k0+kk+2g+{0,1}][m]   (B 4x16 layout)
      v2f b = *(const v2f*)&wrow[k0 + kk + 2 * g];
      acc = __builtin_amdgcn_wmma_f32_16x16x4_f32(false, a, false, b,
                                                  (short)0, acc, false, false);
    }
    __builtin_amdgcn_wave_barrier();   // order loads vs. next chunk's stores
  }

  // epilogue: C/D layout -> row = j + 8*g, col = m
  float bias = b1[m];
#pragma unroll
  for (int j = 0; j < 8; ++j) {
    size_t rr = rowbase + j + 8 * g;
    XW[rr * HD + m] = acc[j] + bias;
  }
}

// ---------------------------------------------------------------------------
// SPMM layer1: h1[dst][j] += w * xw1[src][j], j in [0,16)
// one lane per (edge, col); native global_atomic_add_f32 into L2-resident array
// ---------------------------------------------------------------------------
__global__ __launch_bounds__(256) void spmm16_kernel(const int* __restrict__ src,
                                                     const int* __restrict__ dst,
                                                     const float* __restrict__ w,
                                                     const float* __restrict__ x,
                                                     float* __restrict__ y,
                                                     int n_edges) {
  int idx = blockIdx.x * 256 + threadIdx.x;
  int e = idx >> 4;
  if (e >= n_edges) return;
  int j = idx & 15;
  float v = w[e] * x[(size_t)src[e] * HD + j];
  unsafeAtomicAdd(&y[(size_t)dst[e] * HD + j], v);
}

// ---------------------------------------------------------------------------
// GEMM2: XW2[n][0..6] = (relu(h1[n][:]) * M2[n][:]) @ W2[16][7] + b2
// K=16 -> 4 WMMAs per 16-row tile; B cols 7..15 zero-padded; stride-8 output
// ---------------------------------------------------------------------------
__global__ __launch_bounds__(256) void gemm2_kernel(const float* __restrict__ H,
                                                    const float* __restrict__ M2,
                                                    const float* __restrict__ W2,
                                                    const float* __restrict__ b2,
                                                    float* __restrict__ XW2) {
  const int t    = threadIdx.x;
  const int wave = t >> 5;
  const int lane = t & 31;

  int tile = blockIdx.x * 8 + wave;
  if (tile >= NTILE) tile = NTILE - 1;

  const int g = lane >> 4;
  const int m = lane & 15;
  const size_t row = (size_t)tile * 16 + m;

  v8f acc = {};
#pragma unroll
  for (int kk = 0; kk < HD; kk += 4) {
    int k = kk + 2 * g;
    float2 hv = *(const float2*)&H[row * HD + k];
    float2 mv = *(const float2*)&M2[row * HD + k];
    v2f a, b;
    a.x = fmaxf(hv.x, 0.0f) * mv.x;
    a.y = fmaxf(hv.y, 0.0f) * mv.y;
    b.x = (m < OD) ? W2[k * OD + m]       : 0.0f;
    b.y = (m < OD) ? W2[(k + 1) * OD + m] : 0.0f;
    acc = __builtin_amdgcn_wmma_f32_16x16x4_f32(false, a, false, b,
                                                (short)0, acc, false, false);
  }

  if (m < OD) {
    float bias = b2[m];
#pragma unroll
    for (int j = 0; j < 8; ++j) {
      size_t rr = (size_t)tile * 16 + j + 8 * g;
      XW2[rr * 8 + m] = acc[j] + bias;
    }
  }
}

// ---------------------------------------------------------------------------
// SPMM layer2: out[dst][j] += w * xw2[src][j], j in [0,7); 8 lanes per edge
// ---------------------------------------------------------------------------
__global__ __launch_bounds__(256) void spmm7_kernel(const int* __restrict__ src,
                                                    const int* __restrict__ dst,
                                                    const float* __restrict__ w,
                                                    const float* __restrict__ x,
                                                    float* __restrict__ y,
                                                    int n_edges) {
  int idx = blockIdx.x * 256 + threadIdx.x;
  int e = idx >> 3;
  if (e >= n_edges) return;
  int j = idx & 7;
  if (j < OD) {
    float v = w[e] * x[(size_t)src[e] * 8 + j];
    unsafeAtomicAdd(&y[(size_t)dst[e] * OD + j], v);
  }
}

// ---------------------------------------------------------------------------
extern "C" void kernel_launch(void* const* d_in, const int* in_sizes, int n_in,
                              void* d_out, int out_size, void* d_ws, size_t ws_size,
                              hipStream_t stream) {
  const float* features = (const float*)d_in[0];
  const int*   esrc     = (const int*)  d_in[1];
  const int*   edst     = (const int*)  d_in[2];
  const float* ew       = (const float*)d_in[3];
  const float* mask1    = (const float*)d_in[4];
  const float* mask2    = (const float*)d_in[5];
  const float* W1       = (const float*)d_in[6];
  const float* b1       = (const float*)d_in[7];
  const float* W2       = (const float*)d_in[8];
  const float* b2       = (const float*)d_in[9];
  float* out = (float*)d_out;

  const int n_edges = in_sizes[1];

  // workspace: xw1 [NN*16] | h1 [NN*16]; xw2 aliases xw1 (dead after spmm1)
  float* xw1 = (float*)d_ws;
  float* h1  = xw1 + (size_t)NN * HD;
  float* xw2 = xw1;

  // 1) zero SPMM accumulators (h1 and out)
  {
    int n1 = NN * HD, n2 = NN * OD;
    int nmax = n1 > n2 ? n1 : n2;
    zero_kernel<<<(nmax + 255) / 256, 256, 0, stream>>>(h1, out, n1, n2);
  }
  // 2) dense layer 1 (dropout fused)
  gemm1_kernel<<<(NTILE + 7) / 8, 256, 0, stream>>>(features, mask1, W1, b1, xw1);
  // 3) SPMM layer 1
  {
    long long tot = (long long)n_edges * HD;
    spmm16_kernel<<<(int)((tot + 255) / 256), 256, 0, stream>>>(esrc, edst, ew, xw1, h1, n_edges);
  }
  // 4) relu + dropout + dense layer 2
  gemm2_kernel<<<(NTILE + 7) / 8, 256, 0, stream>>>(h1, mask2, W2, b2, xw2);
  // 5) SPMM layer 2 -> out
  {
    long long tot = (long long)n_edges * 8;
    spmm7_kernel<<<(int)((tot + 255) / 256), 256, 0, stream>>>(esrc, edst, ew, xw2, out, n_edges);
  }
}